// InnerRL_63230508531731
// MI455X (gfx1250) — compile-verified
//
#include <hip/hip_runtime.h>
#include <math.h>

// ---- problem constants (from reference) ----
#define B_   768
#define T_   512
#define N_   5
#define M_   4
#define S_   5.0f
#define P_   61          // N*N + N*M + M*M = 25+20+16
#define PP   64          // padded to 64 columns for WMMA tiling
#define CH   32          // Phi chunk rows staged in LDS
#define NCH  (T_/CH)     // 16 chunks
#define THREADS 256      // 8 waves (wave32)
#define ITERS 10

typedef float v2f __attribute__((ext_vector_type(2)));
typedef float v8f __attribute__((ext_vector_type(8)));

struct Smem {
  float dx[T_][N_];        // 2560
  float du[T_][M_];        // 2048
  float xd[T_][N_];        // 2560
  float Y[T_];             // 512
  float chunk[CH][PP];     // 2048  (Phi rows, padded cols 61..63 = 0)
  float Gm[PP][PP + 1];    // 4160  ([G | c] augmented)
  float c4[4][PP];         // 256
  float w[PP];             // 64
  float K[M_][N_];         // 20
};                          // total 14228 floats = 56912 B < 64KB

// Phi row column c, given per-row dx, du, xdot, udot
__device__ __forceinline__ float phiCol(int c, const float* dx, const float* du,
                                        const float* xd, const float* ud) {
  if (c < 25) {                      // Phi1 = kron(xdot,dx)+kron(dx,xdot)
    int a = c / 5, b = c % 5;
    return xd[a] * dx[b] + dx[a] * xd[b];
  } else if (c < 45) {               // Phi2 = 2*(kron(xdot,du)+kron(dx,udot))
    int c2 = c - 25; int a = c2 >> 2, b = c2 & 3;
    return 2.0f * (xd[a] * du[b] + dx[a] * ud[b]);
  } else if (c < 61) {               // Phi3 = kron(udot,du)+kron(du,udot)
    int c3 = c - 45; int a = c3 >> 2, b = c3 & 3;
    return ud[a] * du[b] + du[a] * ud[b];
  }
  return 0.0f;                       // pad cols 61..63
}

__device__ __forceinline__ void makeUdot(const float K[M_][N_], const float* dx,
                                         const float* du, const float* xd, float* ud) {
#pragma unroll
  for (int m = 0; m < M_; ++m) {
    float kx = 0.f, kxd = 0.f;
#pragma unroll
    for (int n = 0; n < N_; ++n) { kx += K[m][n] * dx[n]; kxd += K[m][n] * xd[n]; }
    ud[m] = -S_ * (du[m] + kx) - kxd;
  }
}

__global__ __launch_bounds__(THREADS)
void innerrl_kernel(const float* __restrict__ x, const float* __restrict__ u,
                    const float* __restrict__ xdot, const float* __restrict__ x0,
                    const float* __restrict__ u0, const float* __restrict__ Qd,
                    const float* __restrict__ Rd, const float* __restrict__ Kinit,
                    float* __restrict__ out) {
  __shared__ Smem s;
  const int b    = blockIdx.x;
  const int tid  = threadIdx.x;
  const int lane = tid & 31;
  const int wv   = tid >> 5;

  const float* xb  = &x[(size_t)b * T_ * N_];
  const float* ub  = &u[(size_t)b * T_ * M_];
  const float* xdb = &xdot[(size_t)b * T_ * N_];

  // gfx1250 prefetch path (global_prefetch_b8) over this block's input slab
  __builtin_prefetch(xb  + tid * 10, 0, 1);
  __builtin_prefetch(ub  + tid * 8,  0, 1);
  __builtin_prefetch(xdb + tid * 10, 0, 1);

  // ---- Phase 0: dx, du, xdot, Y into LDS ----
  float x0r[N_], u0r[M_], Qr[N_], Rr[M_];
#pragma unroll
  for (int n = 0; n < N_; ++n) { x0r[n] = x0[b * N_ + n]; Qr[n] = Qd[n]; }
#pragma unroll
  for (int m = 0; m < M_; ++m) { u0r[m] = u0[b * M_ + m]; Rr[m] = Rd[m]; }

  for (int t = tid; t < T_; t += THREADS) {
    const float* xp  = &xb[t * N_];
    const float* up  = &ub[t * M_];
    const float* xdp = &xdb[t * N_];
    float cost = 0.f;
#pragma unroll
    for (int n = 0; n < N_; ++n) {
      float d = xp[n] - x0r[n]; s.dx[t][n] = d; cost += Qr[n] * d * d;
      s.xd[t][n] = xdp[n];
    }
#pragma unroll
    for (int m = 0; m < M_; ++m) {
      float d = up[m] - u0r[m]; s.du[t][m] = d; cost += Rr[m] * d * d;
    }
    s.Y[t] = expf(-cost) - 1.0f;            // Y = -(1 - exp(-cost))
  }
  if (tid < M_ * N_) s.K[tid / N_][tid % N_] = Kinit[(size_t)b * M_ * N_ + tid];
  __syncthreads();

  // wave's two 16x16 G tiles: same tm, adjacent tn
  const int tm   = (2 * wv) >> 2;
  const int tn0  = (2 * wv) & 3;
  const int tn1  = tn0 + 1;
  const int l    = lane & 15;
  const int half = lane >> 4;

  // ---- 10 outer iterations ----
  for (int it = 0; it < ITERS; ++it) {
    v8f acc0 = {0.f, 0.f, 0.f, 0.f, 0.f, 0.f, 0.f, 0.f};
    v8f acc1 = {0.f, 0.f, 0.f, 0.f, 0.f, 0.f, 0.f, 0.f};
    float csum = 0.f;

    for (int ch = 0; ch < NCH; ++ch) {
      const int base = ch * CH;
      // build Phi chunk: thread -> (row r = tid>>3, col group p = tid&7 -> 8 cols)
      {
        const int r = tid >> 3;
        const int p = tid & 7;
        const int t = base + r;
        float dxr[N_], dur[M_], xdr[N_], ud[M_];
#pragma unroll
        for (int n = 0; n < N_; ++n) { dxr[n] = s.dx[t][n]; xdr[n] = s.xd[t][n]; }
#pragma unroll
        for (int m = 0; m < M_; ++m) { dur[m] = s.du[t][m]; }
        makeUdot(s.K, dxr, dur, xdr, ud);
#pragma unroll
        for (int cc = 0; cc < 8; ++cc)
          s.chunk[r][p * 8 + cc] = phiCol(p * 8 + cc, dxr, dur, xdr, ud);
      }
      __syncthreads();

      // G += Phi_chunk^T * Phi_chunk via V_WMMA_F32_16X16X4_F32
      // A(16x4): v0 = K(2*half), v1 = K(2*half+1), lanes%16 = M; B mirrors with N.
      for (int kb = 0; kb < CH; kb += 4) {
        v2f a, b0, b1;
        a.x  = s.chunk[kb + 2 * half + 0][tm * 16 + l];
        a.y  = s.chunk[kb + 2 * half + 1][tm * 16 + l];
        b0.x = s.chunk[kb + 2 * half + 0][tn0 * 16 + l];
        b0.y = s.chunk[kb + 2 * half + 1][tn0 * 16 + l];
        b1.x = s.chunk[kb + 2 * half + 0][tn1 * 16 + l];
        b1.y = s.chunk[kb + 2 * half + 1][tn1 * 16 + l];
        acc0 = __builtin_amdgcn_wmma_f32_16x16x4_f32(false, a, false, b0,
                                                     (short)0, acc0, false, false);
        acc1 = __builtin_amdgcn_wmma_f32_16x16x4_f32(false, a, false, b1,
                                                     (short)0, acc1, false, false);
      }

      // c += Phi_chunk^T * Y partials: thread -> (m = tid&63, row group q = tid>>6)
      {
        const int m = tid & 63;
        const int q = tid >> 6;
#pragma unroll
        for (int r = q * 8; r < q * 8 + 8; ++r)
          csum += s.chunk[r][m] * s.Y[base + r];
      }
      __syncthreads();
    }

    // write G tiles (C/D layout: row = v + 8*half, col = l) and c partials
#pragma unroll
    for (int v = 0; v < 8; ++v) {
      s.Gm[tm * 16 + v + 8 * half][tn0 * 16 + l] = acc0[v];
      s.Gm[tm * 16 + v + 8 * half][tn1 * 16 + l] = acc1[v];
    }
    s.c4[tid >> 6][tid & 63] = csum;
    __syncthreads();

    if (tid < PP) {
      s.Gm[tid][PP] = s.c4[0][tid] + s.c4[1][tid] + s.c4[2][tid] + s.c4[3][tid];
      if (tid >= P_) {                       // pad rows -> identity, rhs 0
        for (int j = 0; j <= PP; ++j) s.Gm[tid][j] = 0.f;
        s.Gm[tid][tid] = 1.f;
      } else {
        s.Gm[tid][tid] += 1e-5f;             // ridge (pinv cutoff surrogate)
      }
    }
    __syncthreads();

    // Gauss-Jordan on [G | c]: thread -> (row gi = tid&63, col stride group gg)
    const int gi = tid & 63;
    const int gg = tid >> 6;
    for (int k = 0; k < PP; ++k) {
      float f = 0.f;
      if (gi != k) f = s.Gm[gi][k] / s.Gm[k][k];
      __syncthreads();
      if (gi != k)
        for (int j = gg; j <= PP; j += 4)
          s.Gm[gi][j] -= f * s.Gm[k][j];
      __syncthreads();
    }
    if (tid < PP) s.w[tid] = s.Gm[tid][PP] / s.Gm[tid][tid];
    __syncthreads();

    // K update for iterations 0..8 (final iteration keeps K for pred)
    if (it < ITERS - 1) {
      if (tid == 0) {
        // W3 = transpose(reshape(w[45:61],(4,4))); invert 4x4; W2[a][n]=w[25+n*4+a]
        float A[4][8];
#pragma unroll
        for (int i = 0; i < 4; ++i)
#pragma unroll
          for (int j = 0; j < 4; ++j) {
            A[i][j]     = s.w[45 + j * 4 + i];
            A[i][4 + j] = (i == j) ? 1.f : 0.f;
          }
        for (int k = 0; k < 4; ++k) {
          float ip = 1.0f / A[k][k];
#pragma unroll
          for (int j = 0; j < 8; ++j) A[k][j] *= ip;
#pragma unroll
          for (int i = 0; i < 4; ++i) if (i != k) {
            float f = A[i][k];
#pragma unroll
            for (int j = 0; j < 8; ++j) A[i][j] -= f * A[k][j];
          }
        }
#pragma unroll
        for (int m = 0; m < M_; ++m)
#pragma unroll
          for (int n = 0; n < N_; ++n) {
            float v = 0.f;
#pragma unroll
            for (int a = 0; a < 4; ++a) v += A[m][4 + a] * s.w[25 + n * 4 + a];
            if (v != v || fabsf(v) > 1e30f)   // NaN/Inf guard (deterministic fill)
              v = 0.05f * (float)(m * N_ + n - 9);
            s.K[m][n] = v;
          }
      }
      __syncthreads();
    }
  }

  // ---- pred = Phi(K_prev) @ w, written to out[b*T + t] ----
  for (int t = tid; t < T_; t += THREADS) {
    float dxr[N_], dur[M_], xdr[N_], ud[M_];
#pragma unroll
    for (int n = 0; n < N_; ++n) { dxr[n] = s.dx[t][n]; xdr[n] = s.xd[t][n]; }
#pragma unroll
    for (int m = 0; m < M_; ++m) { dur[m] = s.du[t][m]; }
    makeUdot(s.K, dxr, dur, xdr, ud);
    float acc = 0.f;
    for (int c = 0; c < P_; ++c) acc += phiCol(c, dxr, dur, xdr, ud) * s.w[c];
    out[(size_t)b * T_ + t] = acc;
  }
}

extern "C" void kernel_launch(void* const* d_in, const int* in_sizes, int n_in,
                              void* d_out, int out_size, void* d_ws, size_t ws_size,
                              hipStream_t stream) {
  (void)in_sizes; (void)n_in; (void)d_ws; (void)ws_size; (void)out_size;
  const float* x     = (const float*)d_in[0];
  const float* u     = (const float*)d_in[1];
  const float* xdot  = (const float*)d_in[2];
  const float* x0    = (const float*)d_in[3];
  const float* u0    = (const float*)d_in[4];
  const float* Qd    = (const float*)d_in[5];
  const float* Rd    = (const float*)d_in[6];
  const float* Kinit = (const float*)d_in[7];
  innerrl_kernel<<<dim3(B_), dim3(THREADS), 0, stream>>>(
      x, u, xdot, x0, u0, Qd, Rd, Kinit, (float*)d_out);
}